// SimpleLSTM_19404662243752
// MI455X (gfx1250) — compile-verified
//
#include <hip/hip_runtime.h>
#include <math.h>

typedef __attribute__((ext_vector_type(2))) float v2f;
typedef __attribute__((ext_vector_type(8))) float v8f;

constexpr int kB  = 32;
constexpr int kT  = 8192;
constexpr int kI  = 3;
constexpr int kH  = 64;
constexpr int kG  = 256;   // 4*H
constexpr int kBS = 16;    // timesteps per pipeline block (WMMA M dim)
constexpr int kNB = kT / kBS;  // 512 blocks

__device__ __forceinline__ float fsigmoid(float x) {
    return 1.0f / (1.0f + __expf(-x));
}
__device__ __forceinline__ float ftanh_f(float x) {
    float ax = fabsf(x);
    float e  = __expf(-2.0f * ax);        // in (0,1], never overflows
    float t  = (1.0f - e) / (1.0f + e);
    return copysignf(t, x);
}

__global__ __launch_bounds__(512, 1)
void lstm2_fused(const float* __restrict__ x,
                 const float* __restrict__ w_ih0, const float* __restrict__ w_hh0,
                 const float* __restrict__ b_ih0, const float* __restrict__ b_hh0,
                 const float* __restrict__ w_ih1, const float* __restrict__ w_hh1,
                 const float* __restrict__ b_ih1, const float* __restrict__ b_hh1,
                 const float* __restrict__ w_lin, const float* __restrict__ b_lin,
                 float* __restrict__ out)
{
    __shared__ __align__(16) float h0s[2][kH];       // layer0 hidden state, double buffered
    __shared__ __align__(16) float h1s[2][kH];       // layer1 hidden state, double buffered
    __shared__ __align__(16) float h0blk[kBS][kH];   // layer0 h for current block (WMMA A)
    __shared__ __align__(16) float h1blk[kBS][kH];   // layer1 h for current block (final linear)
    __shared__ __align__(16) float xg1[kBS][kG];     // WMMA result: layer1 input gates
    __shared__ __align__(16) float xbuf[2][kBS * kI];// prefetched x, double buffered
    __shared__ __align__(16) float wlin_s[kH];

    const int b    = blockIdx.x;
    const int tid  = threadIdx.x;
    const int lane = tid & 31;
    const int wave = tid >> 5;
    const bool isL0 = (tid < kG);

    // Gate mapping: the 4 gates (i,f,g,o) of hidden unit u sit in 4 adjacent
    // lanes of one wave -> gate exchange is wave-internal (shfl), no barrier.
    const int u     = (tid & 255) >> 2;   // hidden unit 0..63
    const int gtype = tid & 3;            // 0=i 1=f 2=g 3=o (PyTorch order)
    const int grow  = gtype * kH + u;     // row in w_hh / bias vectors
    const int gbase = lane & ~3;          // first lane of this unit's 4-lane group

    // ---- per-thread recurrent weight row in registers (64 VGPRs) ----
    float wr[kH];
    {
        const float* wsrc = isL0 ? (w_hh0 + grow * kH) : (w_hh1 + grow * kH);
        #pragma unroll
        for (int k = 0; k < kH; ++k) wr[k] = wsrc[k];
    }
    // layer0 input weights + combined bias
    float wi0 = 0.f, wi1 = 0.f, wi2 = 0.f, bias0 = 0.f;
    if (isL0) {
        wi0 = w_ih0[grow * kI + 0];
        wi1 = w_ih0[grow * kI + 1];
        wi2 = w_ih0[grow * kI + 2];
        bias0 = b_ih0[grow] + b_hh0[grow];
    }

    // ---- WMMA B fragments: wave w owns gates [16w, 16w+16) of xg1 ----
    // B is KxN (64x16 slice of w_ih1^T). f32 4x16 fragment layout:
    //   lane L: N = L&15; VGPR0 = B[4k0 + 2*(L>>4)][N], VGPR1 = next K.
    const int nbase = wave * 16;
    const int nlane = lane & 15;
    const int khalf = lane >> 4;
    v2f bfrag[16];
    #pragma unroll
    for (int k0 = 0; k0 < 16; ++k0) {
        const int kk = 4 * k0 + 2 * khalf;
        const int g  = nbase + nlane;
        bfrag[k0].x = w_ih1[g * kH + kk];
        bfrag[k0].y = w_ih1[g * kH + kk + 1];
    }
    const float bias1n = b_ih1[nbase + nlane] + b_hh1[nbase + nlane]; // C init (N = lane&15)

    // ---- init ----
    if (tid < kH) { h0s[0][tid] = 0.0f; h1s[0][tid] = 0.0f; wlin_s[tid] = w_lin[tid]; }
    if (tid < kBS * kI) xbuf[0][tid] = x[(size_t)b * kT * kI + tid];  // block 0 x
    float c_st = 0.0f;   // c-state, kept redundantly in all 4 lanes of a unit group
    const float blin = b_lin[0];
    __syncthreads();

    // ---- pipelined block loop: layer0 on block `blk`, layer1 on block `blk-1` ----
    for (int blk = 0; blk <= kNB; ++blk) {
        const int  xb    = blk & 1;
        const bool l0act = (blk < kNB);
        const bool l1act = (blk >= 1);

        for (int s = 0; s < kBS; ++s) {
            const int ps = s & 1;        // h parity: read [ps], write [ps^1]
            if (isL0) {
                if (l0act) {
                    const float* hp = h0s[ps];
                    float a0 = bias0
                             + xbuf[xb][s * kI + 0] * wi0
                             + xbuf[xb][s * kI + 1] * wi1
                             + xbuf[xb][s * kI + 2] * wi2;
                    float a1 = 0.f, a2 = 0.f, a3 = 0.f;
                    #pragma unroll
                    for (int k = 0; k < kH; k += 4) {
                        a0 += hp[k + 0] * wr[k + 0];
                        a1 += hp[k + 1] * wr[k + 1];
                        a2 += hp[k + 2] * wr[k + 2];
                        a3 += hp[k + 3] * wr[k + 3];
                    }
                    const float acc = (a0 + a2) + (a1 + a3);
                    const float act = (gtype == 2) ? ftanh_f(acc) : fsigmoid(acc);
                    // wave-internal gate exchange (no barrier)
                    const float ig = __shfl(act, gbase + 0, 32);
                    const float fg = __shfl(act, gbase + 1, 32);
                    const float gg = __shfl(act, gbase + 2, 32);
                    const float og = __shfl(act, gbase + 3, 32);
                    c_st = fg * c_st + ig * gg;
                    const float hv = og * ftanh_f(c_st);
                    if (gtype == 0) {
                        h0s[ps ^ 1][u] = hv;
                        h0blk[s][u]    = hv;
                    }
                }
            } else {
                if (l1act) {
                    const float* hp = h1s[ps];
                    float a0 = xg1[s][grow];   // precomputed by WMMA (incl. biases)
                    float a1 = 0.f, a2 = 0.f, a3 = 0.f;
                    #pragma unroll
                    for (int k = 0; k < kH; k += 4) {
                        a0 += hp[k + 0] * wr[k + 0];
                        a1 += hp[k + 1] * wr[k + 1];
                        a2 += hp[k + 2] * wr[k + 2];
                        a3 += hp[k + 3] * wr[k + 3];
                    }
                    const float acc = (a0 + a2) + (a1 + a3);
                    const float act = (gtype == 2) ? ftanh_f(acc) : fsigmoid(acc);
                    const float ig = __shfl(act, gbase + 0, 32);
                    const float fg = __shfl(act, gbase + 1, 32);
                    const float gg = __shfl(act, gbase + 2, 32);
                    const float og = __shfl(act, gbase + 3, 32);
                    c_st = fg * c_st + ig * gg;
                    const float hv = og * ftanh_f(c_st);
                    if (gtype == 0) {
                        h1s[ps ^ 1][u] = hv;
                        h1blk[s][u]    = hv;
                    }
                }
            }
            __syncthreads();   // single barrier per step
        }

        // ================= block-boundary phase =================
        // prefetch next x block (issue loads early, store to LDS at end of phase)
        float xpre = 0.0f;
        const bool doxpre = (blk + 1 < kNB) && (tid < kBS * kI);
        if (doxpre) xpre = x[(size_t)b * kT * kI + (size_t)(blk + 1) * kBS * kI + tid];

        // final linear + skip for layer1's completed block (blk-1):
        // 4 lanes per step, 16 FMAs each, shfl_xor reduction
        if (l1act && tid < 4 * kBS) {
            const int  st = tid >> 2;
            const int  q  = tid & 3;
            float part = 0.0f;
            #pragma unroll
            for (int j = 0; j < 16; ++j)
                part += h1blk[st][q * 16 + j] * wlin_s[q * 16 + j];
            part += __shfl_xor(part, 1, 32);
            part += __shfl_xor(part, 2, 32);
            if (q == 0) {
                const int tp = (blk - 1) * kBS + st;
                out[(size_t)b * kT + tp] =
                    part + blin + x[((size_t)b * kT + tp) * kI];  // + x[...,0] skip
            }
        }

        // WMMA: xg1[s][g] = bias1[g] + sum_k h0blk[s][k] * w_ih1[g][k]
        // A (16 steps x 64) f32 fragment: lane L: M = L&15, K pair = 4k0+2*(L>>4), +1
        if (l0act) {
            v8f c;
            #pragma unroll
            for (int v = 0; v < 8; ++v) c[v] = bias1n;
#if __has_builtin(__builtin_amdgcn_wmma_f32_16x16x4_f32)
            #pragma unroll
            for (int k0 = 0; k0 < 16; ++k0) {
                const int kk = 4 * k0 + 2 * khalf;
                const v2f a = *(const v2f*)&h0blk[nlane][kk];   // ds_load_b64
                c = __builtin_amdgcn_wmma_f32_16x16x4_f32(
                        false, a, false, bfrag[k0], (short)0, c, false, false);
            }
#else
            // VALU fallback: same fragment math, per lane
            #pragma unroll
            for (int v = 0; v < 8; ++v) {
                const int m = v + 8 * khalf;
                float acc = 0.0f;
                for (int k = 0; k < kH; ++k)
                    acc += h0blk[m][k] * w_ih1[(nbase + nlane) * kH + k];
                c[v] += acc;
            }
#endif
            // C layout: reg v, lane L -> M = v + 8*(L>>4), N = nbase + (L&15)
            #pragma unroll
            for (int v = 0; v < 8; ++v)
                xg1[v + 8 * khalf][nbase + nlane] = c[v];
        }

        if (doxpre) xbuf[xb ^ 1][tid] = xpre;
        __syncthreads();
    }
}

extern "C" void kernel_launch(void* const* d_in, const int* in_sizes, int n_in,
                              void* d_out, int out_size, void* d_ws, size_t ws_size,
                              hipStream_t stream) {
    const float* x     = (const float*)d_in[0];
    const float* w_ih0 = (const float*)d_in[1];
    const float* w_hh0 = (const float*)d_in[2];
    const float* b_ih0 = (const float*)d_in[3];
    const float* b_hh0 = (const float*)d_in[4];
    const float* w_ih1 = (const float*)d_in[5];
    const float* w_hh1 = (const float*)d_in[6];
    const float* b_ih1 = (const float*)d_in[7];
    const float* b_hh1 = (const float*)d_in[8];
    const float* w_lin = (const float*)d_in[9];
    const float* b_lin = (const float*)d_in[10];
    float* out = (float*)d_out;

    lstm2_fused<<<dim3(kB), dim3(512), 0, stream>>>(
        x, w_ih0, w_hh0, b_ih0, b_hh0,
        w_ih1, w_hh1, b_ih1, b_hh1,
        w_lin, b_lin, out);
}